// StarE_28252294873230
// MI455X (gfx1250) — compile-verified
//
#include <hip/hip_runtime.h>
#include <hip/hip_bf16.h>
#include <math.h>

#define N_ENT  100000
#define N_RELS 200
#define DIM    200
#define HALF_D 100
#define N_EDGE 1000000
#define MT     8          // M tiles per block (one per wave)

typedef __attribute__((ext_vector_type(2))) float v2f;
typedef __attribute__((ext_vector_type(8))) float v8f;

// ---------------------------------------------------------------- degrees
__global__ __launch_bounds__(256) void degree_kernel(const int* __restrict__ src,
                                                     const int* __restrict__ dst,
                                                     float* __restrict__ deg_in,
                                                     float* __restrict__ deg_out) {
    int e = blockIdx.x * 256 + threadIdx.x;
    if (e >= N_EDGE) return;
    atomicAdd(deg_in  + dst[e], 1.0f);   // in-direction: deg over dst
    atomicAdd(deg_out + src[e], 1.0f);   // out-direction: deg over src
}

__global__ __launch_bounds__(256) void inv_sqrt_kernel(float* __restrict__ a,
                                                       float* __restrict__ b) {
    int i = blockIdx.x * 256 + threadIdx.x;
    if (i >= N_ENT) return;
    float d = a[i]; a[i] = (d > 0.0f) ? rsqrtf(d) : 0.0f;
    float e = b[i]; b[i] = (e > 0.0f) ? rsqrtf(e) : 0.0f;
}

// Branchless complex-rotation element:  out[j] of rotate(h, r)
//  j <  100: h[j]*r[j]      - h[j+100]*r[j+100]
//  j >= 100: h[j]*r[j-100]  + h[j-100]*r[j]
__device__ __forceinline__ float rot_elem(const float* __restrict__ h,
                                          const float* __restrict__ r, int j) {
    const bool lo = (j < HALF_D);
    const int  jo = lo ? j + HALF_D : j - HALF_D;
    const float s = lo ? -1.0f : 1.0f;
    const float hx = h[j],            hy = h[jo];
    const float rx = r[lo ? j : jo],  ry = r[lo ? jo : j];
    return fmaf(s * hy, ry, hx * rx);
}

// ------------------------------------------- per-edge rotate + scatter-add
// One wave32 per edge; lanes stripe the 200-dim embedding. Both directions
// fused so each ent row is gathered once per edge endpoint.
__global__ __launch_bounds__(256) void edge_scatter_kernel(
    const float* __restrict__ ent, const float* __restrict__ rel,
    const int* __restrict__ src, const int* __restrict__ dst,
    const int* __restrict__ et,
    const float* __restrict__ inv_in, const float* __restrict__ inv_out,
    float* __restrict__ acc_in, float* __restrict__ acc_out) {
    const int lane = threadIdx.x & 31;
    const int wave = threadIdx.x >> 5;
    const long e = (long)blockIdx.x * 8 + wave;
    if (e >= N_EDGE) return;
    const int s = src[e], d = dst[e], t = et[e];
    const float nin  = inv_in[d]  * inv_in[s];
    const float nout = inv_out[s] * inv_out[d];
    const float* __restrict__ hs   = ent + (long)s * DIM;
    const float* __restrict__ hd   = ent + (long)d * DIM;
    const float* __restrict__ rin  = rel + (long)t * DIM;            // 320 KB table lives in L2
    const float* __restrict__ rout = rel + (long)(t + N_RELS) * DIM;
    float* __restrict__ ain  = acc_in  + (long)d * DIM;
    float* __restrict__ aout = acc_out + (long)s * DIM;
    for (int j = lane; j < DIM; j += 32) {
        atomicAdd(ain + j,  nin  * rot_elem(hs, rin,  j));
        atomicAdd(aout + j, nout * rot_elem(hd, rout, j));
    }
}

// ------------------------------------------------ fused 3-way GEMM (WMMA)
// out = (acc_in@Win + acc_out@Wout + rotate(ent,loop)@Wloop) / 3 + bias
// 8 waves per block, each owning one 16x16 output tile of the same N tile.
// B tiles (16 columns of each weight matrix, all K) staged zero-padded in
// LDS once per block, so the K loop has no masked loads / exec churn around
// the V_WMMA_F32_16X16X4_F32 chain.
__global__ __launch_bounds__(256) void gemm_fused_kernel(
    const float* __restrict__ Ain, const float* __restrict__ Aout,
    const float* __restrict__ ent, const float* __restrict__ lrel,
    const float* __restrict__ Win, const float* __restrict__ Wout,
    const float* __restrict__ Wloop, const float* __restrict__ bias,
    float* __restrict__ out) {
    __shared__ float Bsh[3][DIM][16];            // 37.5 KB of 320 KB WGP LDS

    const int tid    = threadIdx.x;
    const int lane   = tid & 31;
    const int wave   = tid >> 5;
    const int n_base = blockIdx.x * 16;

    // ---- stage B tiles (masking confined to this prologue)
    for (int idx = tid; idx < DIM * 16; idx += 256) {
        const int kk = idx >> 4, nc = idx & 15;
        const int col = n_base + nc;
        const bool ok = (col < DIM);
        Bsh[0][kk][nc] = ok ? Win[(long)kk * DIM + col]   : 0.0f;
        Bsh[1][kk][nc] = ok ? Wout[(long)kk * DIM + col]  : 0.0f;
        Bsh[2][kk][nc] = ok ? Wloop[(long)kk * DIM + col] : 0.0f;
    }
    __syncthreads();

    // ---- per-wave tile coordinates
    const int m_base = (blockIdx.y * MT + wave) * 16;
    int m = m_base + (lane & 15);
    if (m >= N_ENT) m = N_ENT - 1;               // clamp: loads stay in bounds
    const int khalf = (lane & 16) ? 2 : 0;       // A: upper half-wave = K+2,K+3
    const int nc    = lane & 15;                 // B/C: N within tile

    const float* __restrict__ rowIn  = Ain  + (long)m * DIM;
    const float* __restrict__ rowOut = Aout + (long)m * DIM;
    const float* __restrict__ rowEnt = ent  + (long)m * DIM;

    v8f c = {};
#pragma unroll 2
    for (int k = 0; k < DIM; k += 4) {
        const int kk = k + khalf;
        v2f a0, a1, a2, b0, b1, b2;
        a0 = *(const v2f*)(rowIn + kk);          // kk even, 8B-aligned
        a1 = *(const v2f*)(rowOut + kk);
        a2.x = rot_elem(rowEnt, lrel, kk);
        a2.y = rot_elem(rowEnt, lrel, kk + 1);
        b0.x = Bsh[0][kk][nc];  b0.y = Bsh[0][kk + 1][nc];
        b1.x = Bsh[1][kk][nc];  b1.y = Bsh[1][kk + 1][nc];
        b2.x = Bsh[2][kk][nc];  b2.y = Bsh[2][kk + 1][nc];
        c = __builtin_amdgcn_wmma_f32_16x16x4_f32(false, a0, false, b0, (short)0, c, false, false);
        c = __builtin_amdgcn_wmma_f32_16x16x4_f32(false, a1, false, b1, (short)0, c, false, false);
        c = __builtin_amdgcn_wmma_f32_16x16x4_f32(false, a2, false, b2, (short)0, c, false, false);
    }

    // ---- epilogue: C/D layout -> row = m_base + j (+8 for upper half-wave)
    const int ncol = n_base + nc;
    if (ncol < DIM) {
        const int rbase = m_base + ((lane & 16) ? 8 : 0);
        const float bv = bias[ncol];
#pragma unroll
        for (int j = 0; j < 8; ++j) {
            const int row = rbase + j;
            if (row < N_ENT)
                out[(long)row * DIM + ncol] = c[j] * (1.0f / 3.0f) + bv;
        }
    }
}

// ------------------------------------------------------------- batch-norm
__global__ __launch_bounds__(256) void bn_stats_kernel(const float* __restrict__ x,
                                                       float* __restrict__ mean,
                                                       float* __restrict__ var) {
    const int col = blockIdx.x;
    float s = 0.0f, s2 = 0.0f;
    for (int r = threadIdx.x; r < N_ENT; r += 256) {
        const float v = x[(long)r * DIM + col];
        s += v; s2 += v * v;
    }
    __shared__ float sh[256], sh2[256];
    sh[threadIdx.x] = s; sh2[threadIdx.x] = s2;
    __syncthreads();
    for (int o = 128; o > 0; o >>= 1) {
        if (threadIdx.x < o) {
            sh[threadIdx.x]  += sh[threadIdx.x + o];
            sh2[threadIdx.x] += sh2[threadIdx.x + o];
        }
        __syncthreads();
    }
    if (threadIdx.x == 0) {
        const float m = sh[0] * (1.0f / N_ENT);
        mean[col] = m;
        var[col]  = sh2[0] * (1.0f / N_ENT) - m * m;   // population var (matches jnp)
    }
}

__global__ __launch_bounds__(256) void bn_tanh_kernel(float* __restrict__ y,
                                                      const float* __restrict__ mean,
                                                      const float* __restrict__ var,
                                                      const float* __restrict__ gamma,
                                                      const float* __restrict__ beta) {
    const long i = (long)blockIdx.x * 256 + threadIdx.x;
    if (i >= (long)N_ENT * DIM) return;
    const int c = (int)(i % DIM);
    const float v = (y[i] - mean[c]) * rsqrtf(var[c] + 1e-5f) * gamma[c] + beta[c];
    y[i] = tanhf(v);
}

// ---------------------------------------------------------------- launch
extern "C" void kernel_launch(void* const* d_in, const int* in_sizes, int n_in,
                              void* d_out, int out_size, void* d_ws, size_t ws_size,
                              hipStream_t stream) {
    const float* ent    = (const float*)d_in[0];
    const float* rel    = (const float*)d_in[1];
    const float* lrel   = (const float*)d_in[2];
    const float* w_in   = (const float*)d_in[3];
    const float* w_out  = (const float*)d_in[4];
    const float* w_loop = (const float*)d_in[5];
    const float* bias   = (const float*)d_in[6];
    const float* gamma  = (const float*)d_in[7];
    const float* beta   = (const float*)d_in[8];
    const int*   eidx   = (const int*)d_in[9];
    const int*   etype  = (const int*)d_in[10];
    const int* src = eidx;
    const int* dst = eidx + N_EDGE;
    float* out = (float*)d_out;

    // workspace layout (floats): acc_in[20M] acc_out[20M] deg_in[100k] deg_out[100k] mean[200] var[200]
    float* acc_in  = (float*)d_ws;
    float* acc_out = acc_in  + (size_t)N_ENT * DIM;
    float* deg_in  = acc_out + (size_t)N_ENT * DIM;
    float* deg_out = deg_in  + N_ENT;
    float* mean    = deg_out + N_ENT;
    float* var     = mean + DIM;

    hipMemsetAsync(acc_in, 0, sizeof(float) * 2 * (size_t)N_ENT * DIM, stream);
    hipMemsetAsync(deg_in, 0, sizeof(float) * 2 * (size_t)N_ENT, stream);

    degree_kernel<<<(N_EDGE + 255) / 256, 256, 0, stream>>>(src, dst, deg_in, deg_out);
    inv_sqrt_kernel<<<(N_ENT + 255) / 256, 256, 0, stream>>>(deg_in, deg_out);
    edge_scatter_kernel<<<N_EDGE / 8, 256, 0, stream>>>(ent, rel, src, dst, etype,
                                                        deg_in, deg_out, acc_in, acc_out);
    const int m_tiles = (N_ENT + 15) / 16;                    // 6250
    dim3 gg((DIM + 15) / 16, (m_tiles + MT - 1) / MT);        // 13 x 782
    gemm_fused_kernel<<<gg, 256, 0, stream>>>(acc_in, acc_out, ent, lrel,
                                              w_in, w_out, w_loop, bias, out);
    bn_stats_kernel<<<DIM, 256, 0, stream>>>(out, mean, var);
    const long total = (long)N_ENT * DIM;
    bn_tanh_kernel<<<(unsigned)((total + 255) / 256), 256, 0, stream>>>(out, mean, var, gamma, beta);
}